// VolumeNormalizer_4818953307046
// MI455X (gfx1250) — compile-verified
//
#include <hip/hip_runtime.h>
#include <hip/hip_bf16.h>

// ---------------------------------------------------------------------------
// VolumeNormalizer on gfx1250 (MI455X).
// Hot loop: c(16x4000) = d(16x16000) @ Vf^T(16000x4000)  via V_WMMA_F32_16X16X4_F32.
// Memory-bound: 256 MB of Vf streamed per axis pass, 3 passes (~11 us/pass at
// 23.3 TB/s). WMMA matmul is register-pipelined: next 16x128 tile is loaded
// into VGPRs while the current LDS tile feeds the WMMAs.
// ---------------------------------------------------------------------------

#define NB     16
#define NPTS   6000
#define NTRI   16000
#define NMOV   4000
#define NBND   1500
#define KTILE  128
#define LSTR   132          // KTILE + 4 LDS pad -> conflict-free b64 fragment reads
#define KSPLIT 5            // 16000 / 5 = 3200 = 25 * KTILE per block

typedef __attribute__((ext_vector_type(2))) float v2f;
typedef __attribute__((ext_vector_type(8))) float v8f;

// ---------------- init: pz[b] = points_zero ; pts(out) = x ------------------
__global__ void vn_init_pz(const float* __restrict__ p0, const float* __restrict__ x,
                           float* __restrict__ pz, float* __restrict__ pts) {
    int i = blockIdx.x * blockDim.x + threadIdx.x;
    if (i >= NB * NPTS) return;
    int b = i / NPTS, p = i % NPTS;
    float* dst = pz + ((size_t)b * NPTS + p) * 3;
    dst[0] = p0[p * 3 + 0];
    dst[1] = p0[p * 3 + 1];
    dst[2] = p0[p * 3 + 2];
    if (p < NMOV) {
        const float* xs = x + ((size_t)b * NMOV + p) * 3;
        float* pd = pts + ((size_t)b * NMOV + p) * 3;
        pd[0] = xs[0]; pd[1] = xs[1]; pd[2] = xs[2];
    }
}

// ---------------- scatter boundary y into pz (coords 0 and 2) ---------------
__global__ void vn_scatter_y(const float* __restrict__ y, const int* __restrict__ idx2,
                             float* __restrict__ pz) {
    int i = blockIdx.x * blockDim.x + threadIdx.x;
    if (i >= NB * NBND) return;
    int b = i / NBND, j = i % NBND;
    int p = idx2[j];
    float* dst = pz + ((size_t)b * NPTS + p) * 3;
    dst[0] = y[(size_t)b * (2 * NBND) + 2 * j + 0];
    dst[2] = y[(size_t)b * (2 * NBND) + 2 * j + 1];
}

// ---------------- scatter moving pts into pz (all 3 coords) -----------------
__global__ void vn_scatter_pts(const float* __restrict__ pts, const int* __restrict__ idx1,
                               float* __restrict__ pz) {
    int i = blockIdx.x * blockDim.x + threadIdx.x;
    if (i >= NB * NMOV) return;
    int b = i / NMOV, n = i % NMOV;
    int p = idx1[n];
    const float* src = pts + ((size_t)b * NMOV + n) * 3;
    float* dst = pz + ((size_t)b * NPTS + p) * 3;
    dst[0] = src[0]; dst[1] = src[1]; dst[2] = src[2];
}

// ---------------- zero helper ----------------------------------------------
__global__ void vn_zero(float* __restrict__ p, int n) {
    int i = blockIdx.x * blockDim.x + threadIdx.x;
    if (i < n) p[i] = 0.0f;
}

// ---------------- volumes: accum[0]=vol_const, accum[1+b]=vol(pz[b]) --------
__global__ __launch_bounds__(256) void vn_volume(const float* __restrict__ p0,
                                                 const float* __restrict__ pz,
                                                 const int* __restrict__ tris,
                                                 float* __restrict__ accum) {
    __shared__ float red[256];
    int t = blockIdx.x * 256 + threadIdx.x;
    float vbase = 0.0f;
    float vb[NB];
#pragma unroll
    for (int b = 0; b < NB; ++b) vb[b] = 0.0f;
    if (t < NTRI) {
        int i0 = tris[3 * t + 0], i1 = tris[3 * t + 1], i2 = tris[3 * t + 2];
        {
            const float* P = p0;
            float y0 = P[i0 * 3 + 1], z0 = P[i0 * 3 + 2];
            float det = (P[i1 * 3 + 1] - y0) * (P[i2 * 3 + 2] - z0)
                      - (P[i1 * 3 + 2] - z0) * (P[i2 * 3 + 1] - y0);
            vbase = (P[i0 * 3] + P[i1 * 3] + P[i2 * 3]) * det * (1.0f / 6.0f);
        }
        for (int b = 0; b < NB; ++b) {
            const float* P = pz + (size_t)b * NPTS * 3;
            float y0 = P[i0 * 3 + 1], z0 = P[i0 * 3 + 2];
            float det = (P[i1 * 3 + 1] - y0) * (P[i2 * 3 + 2] - z0)
                      - (P[i1 * 3 + 2] - z0) * (P[i2 * 3 + 1] - y0);
            vb[b] = (P[i0 * 3] + P[i1 * 3] + P[i2 * 3]) * det * (1.0f / 6.0f);
        }
    }
    for (int ch = 0; ch < NB + 1; ++ch) {
        float v = (ch == 0) ? vbase : vb[ch - 1];
        red[threadIdx.x] = v;
        __syncthreads();
        for (int s = 128; s > 0; s >>= 1) {
            if (threadIdx.x < s) red[threadIdx.x] += red[threadIdx.x + s];
            __syncthreads();
        }
        if (threadIdx.x == 0) atomicAdd(&accum[ch], red[0]);
        __syncthreads();
    }
}

__global__ void vn_compute_a(const float* __restrict__ accum, float* __restrict__ a) {
    int i = threadIdx.x;
    if (i < NB) a[i] = (accum[0] - accum[1 + i]) * (1.0f / 3.0f);
}

// ---------------- per-axis triangle determinants: d[b][t] -------------------
__global__ void vn_dets(const float* __restrict__ pz, const int* __restrict__ tris,
                        float* __restrict__ dbuf, int axis) {
    int t = blockIdx.x * blockDim.x + threadIdx.x;
    if (t >= NTRI) return;
    int i0 = tris[3 * t + 0], i1 = tris[3 * t + 1], i2 = tris[3 * t + 2];
    for (int b = 0; b < NB; ++b) {
        const float* P = pz + (size_t)b * NPTS * 3;
        float a00, a01, a10, a11;
        if (axis == 2) {               // coords (x,y), pivot vertex 2
            float px = P[i2 * 3 + 0], py = P[i2 * 3 + 1];
            a00 = P[i0 * 3 + 0] - px;  a01 = P[i0 * 3 + 1] - py;
            a10 = P[i1 * 3 + 0] - px;  a11 = P[i1 * 3 + 1] - py;
        } else if (axis == 1) {        // coords (x,z), pivot vertex 1
            float px = P[i1 * 3 + 0], pzv = P[i1 * 3 + 2];
            a00 = P[i0 * 3 + 0] - px;  a01 = P[i0 * 3 + 2] - pzv;
            a10 = P[i2 * 3 + 0] - px;  a11 = P[i2 * 3 + 2] - pzv;
        } else {                       // coords (y,z), pivot vertex 0
            float py = P[i0 * 3 + 1], pzv = P[i0 * 3 + 2];
            a00 = P[i1 * 3 + 1] - py;  a01 = P[i1 * 3 + 2] - pzv;
            a10 = P[i2 * 3 + 1] - py;  a11 = P[i2 * 3 + 2] - pzv;
        }
        dbuf[(size_t)b * NTRI + t] = (a00 * a11 - a01 * a10) * (1.0f / 6.0f);
    }
}

// ---------------- WMMA matmul: c(16xNMOV) += d(16xNTRI) @ Vf^T --------------
// grid.x = (NMOV/16) * KSPLIT, block = 256 threads (8 waves).
// Register-pipelined: while the current 16x128 LDS tile feeds 4 WMMAs/wave,
// the next tile's global_load_b128s are already in flight into VGPRs.
__global__ __launch_bounds__(256) void vn_coeff_wmma(const float* __restrict__ Vf,
                                                     const float* __restrict__ dbuf,
                                                     float* __restrict__ cbuf) {
    __shared__ float sVf[16 * LSTR];
    __shared__ float sD[16 * LSTR];
    __shared__ float sRed[8 * 256];

    const int ntiles = NMOV / 16;                 // 250
    const int ntile = blockIdx.x % ntiles;
    const int kseg = blockIdx.x / ntiles;
    const int kseg_len = NTRI / KSPLIT;           // 3200
    const int kbase0 = kseg * kseg_len;
    const int N0 = ntile * 16;

    const int tid = threadIdx.x;
    const int lane = tid & 31;
    const int wave = tid >> 5;
    const int row = tid >> 4;                     // 0..15 (tile row for loads)
    const int col = (tid & 15) * 8;               // 0..120

    const int m = lane & 15;                      // fragment row (M or N)
    const int koff = (lane >> 4) * 2;             // lanes 16-31 hold K+2,K+3

    const float4* gv = (const float4*)(Vf + (size_t)(N0 + row) * NTRI + kbase0 + col);
    const float4* gd = (const float4*)(dbuf + (size_t)row * NTRI + kbase0 + col);
    float* pv = &sVf[row * LSTR + col];
    float* pd = &sD[row * LSTR + col];

    v8f acc = {};

    const int NT = kseg_len / KTILE;              // 25 tiles per block
    // prologue: tile 0 into registers
    float4 rv0 = gv[0], rv1 = gv[1];
    float4 rd0 = gd[0], rd1 = gd[1];

    for (int it = 0; it < NT; ++it) {
        // registers -> LDS
        ((float4*)pv)[0] = rv0; ((float4*)pv)[1] = rv1;
        ((float4*)pd)[0] = rd0; ((float4*)pd)[1] = rd1;
        __syncthreads();

        // issue next tile's global loads now; latency hides under the WMMAs
        if (it + 1 < NT) {
            const float4* gvn = gv + (size_t)(it + 1) * (KTILE / 4);
            const float4* gdn = gd + (size_t)(it + 1) * (KTILE / 4);
            rv0 = gvn[0]; rv1 = gvn[1];
            rd0 = gdn[0]; rd1 = gdn[1];
            if (it + 2 < NT)
                __builtin_prefetch((const float*)(gv + (size_t)(it + 2) * (KTILE / 4)), 0, 3);
        }

        // constant trip count: 4 WMMAs per wave, no EXEC games
#pragma unroll
        for (int i = 0; i < 4; ++i) {
            const int kt = wave * 4 + i * 32;
            v2f afrag = *(const v2f*)&sD[m * LSTR + kt + koff];
            v2f bfrag = *(const v2f*)&sVf[m * LSTR + kt + koff];
            acc = __builtin_amdgcn_wmma_f32_16x16x4_f32(
                false, afrag, false, bfrag, (short)0, acc, false, false);
        }
        __syncthreads();
    }

    // cross-wave reduction of the 16x16 accumulator tiles
#pragma unroll
    for (int r = 0; r < 8; ++r) sRed[wave * 256 + r * 32 + lane] = acc[r];
    __syncthreads();
    if (wave == 0) {
        const int N = lane & 15;
        const int Mhi = (lane >> 4) * 8;
#pragma unroll
        for (int r = 0; r < 8; ++r) {
            float sum = 0.0f;
#pragma unroll
            for (int w = 0; w < 8; ++w) sum += sRed[w * 256 + r * 32 + lane];
            int M = r + Mhi;                       // C/D layout: M = r + 8*(lane>=16)
            atomicAdd(&cbuf[(size_t)M * NMOV + N0 + N], sum);
        }
    }
}

// ---------------- s[b] = sum_n c[b][n]^2 (one block per batch) --------------
__global__ __launch_bounds__(256) void vn_sumsq(const float* __restrict__ c,
                                                float* __restrict__ s) {
    __shared__ float red[256];
    int b = blockIdx.x;
    float acc = 0.0f;
    for (int n = threadIdx.x; n < NMOV; n += 256) {
        float v = c[(size_t)b * NMOV + n];
        acc += v * v;
    }
    red[threadIdx.x] = acc;
    __syncthreads();
    for (int st = 128; st > 0; st >>= 1) {
        if (threadIdx.x < st) red[threadIdx.x] += red[threadIdx.x + st];
        __syncthreads();
    }
    if (threadIdx.x == 0) s[b] = red[0];
}

// ---------------- pts[:, :, axis] += c * a/(s+eps); re-scatter into pz ------
__global__ void vn_update(float* __restrict__ pts, float* __restrict__ pz,
                          const float* __restrict__ c, const float* __restrict__ a,
                          const float* __restrict__ s, const int* __restrict__ idx1,
                          int axis) {
    int i = blockIdx.x * blockDim.x + threadIdx.x;
    if (i >= NB * NMOV) return;
    int b = i / NMOV, n = i % NMOV;
    float scale = a[b] / (s[b] + 1e-8f);
    float* pp = pts + ((size_t)b * NMOV + n) * 3;
    pp[axis] += c[(size_t)b * NMOV + n] * scale;
    int p = idx1[n];
    float* dst = pz + ((size_t)b * NPTS + p) * 3;
    dst[0] = pp[0]; dst[1] = pp[1]; dst[2] = pp[2];
}

// ---------------------------------------------------------------------------
extern "C" void kernel_launch(void* const* d_in, const int* in_sizes, int n_in,
                              void* d_out, int out_size, void* d_ws, size_t ws_size,
                              hipStream_t stream) {
    const float* x   = (const float*)d_in[0];
    const float* y   = (const float*)d_in[1];
    const float* p0  = (const float*)d_in[2];
    const float* Vf  = (const float*)d_in[3];
    const int* idx1  = (const int*)d_in[4];
    const int* idx2  = (const int*)d_in[5];
    const int* tris  = (const int*)d_in[6];
    float* pts = (float*)d_out;                   // (16, 4000, 3) float32

    float* ws    = (float*)d_ws;
    float* pz    = ws;                            // 16*6000*3 = 288000
    float* dbuf  = ws + 288000;                   // 16*16000  = 256000
    float* cbuf  = ws + 544000;                   // 16*4000   =  64000
    float* svals = ws + 608000;                   // 16 (contiguous after cbuf)
    float* accum = ws + 608016;                   // 17
    float* avals = ws + 608048;                   // 16

    // build pz and pts (launch order gives indices_2 -> indices_1 precedence)
    vn_init_pz<<<(NB * NPTS + 255) / 256, 256, 0, stream>>>(p0, x, pz, pts);
    vn_scatter_y<<<(NB * NBND + 255) / 256, 256, 0, stream>>>(y, idx2, pz);
    vn_scatter_pts<<<(NB * NMOV + 255) / 256, 256, 0, stream>>>(pts, idx1, pz);

    // a[b] = (vol_const - vol(pz[b])) / 3
    vn_zero<<<1, 32, 0, stream>>>(accum, NB + 1);
    vn_volume<<<(NTRI + 255) / 256, 256, 0, stream>>>(p0, pz, tris, accum);
    vn_compute_a<<<1, 32, 0, stream>>>(accum, avals);

    const int axes[3] = {2, 1, 0};
    for (int it = 0; it < 3; ++it) {
        int axis = axes[it];
        vn_zero<<<(64016 + 255) / 256, 256, 0, stream>>>(cbuf, 64016);  // cbuf + svals
        vn_dets<<<(NTRI + 255) / 256, 256, 0, stream>>>(pz, tris, dbuf, axis);
        vn_coeff_wmma<<<(NMOV / 16) * KSPLIT, 256, 0, stream>>>(Vf, dbuf, cbuf);
        vn_sumsq<<<NB, 256, 0, stream>>>(cbuf, svals);
        vn_update<<<(NB * NMOV + 255) / 256, 256, 0, stream>>>(pts, pz, cbuf, avals,
                                                               svals, idx1, axis);
    }
}